// CostSensitiveCrossEntropyLossN_57037165690984
// MI455X (gfx1250) — compile-verified
//
#include <hip/hip_runtime.h>

#define WAVES_PER_BLOCK 8
#define ROWS_PER_WAVE   8
#define KMAX            32     // per-lane floats per row (supports C <= 1024)
#define LDS_ROW_FLOATS  1024   // row buffer (C <= 1024)

// ---- CDNA5 async global->LDS data movement --------------------------------
// For flat/shared pointers the low 32 bits are the wave-relative LDS address
// (ISA 10.2); async VDST uses the same convention.
__device__ __forceinline__ unsigned lds_addr_u32(const void* p) {
    return (unsigned)(unsigned long long)p;
}
__device__ __forceinline__ void async_copy_b128(unsigned lds_byte_addr, unsigned long long gaddr) {
    asm volatile("global_load_async_to_lds_b128 %0, %1, off"
                 :: "v"(lds_byte_addr), "v"(gaddr)
                 : "memory");
}
__device__ __forceinline__ void wait_async_0() { asm volatile("s_wait_asynccnt 0x0" ::: "memory"); }
__device__ __forceinline__ void wait_async_8() { asm volatile("s_wait_asynccnt 0x8" ::: "memory"); }

// ---------------------------------------------------------------------------
__global__ void phase0_init(unsigned* __restrict__ counts, long long nC2,
                            unsigned* __restrict__ minbits, unsigned* __restrict__ maxbits) {
    long long i = (long long)blockIdx.x * blockDim.x + threadIdx.x;
    if (i < nC2) counts[i] = 0u;
    if (i == 0) { *minbits = 0x7F800000u; /* +inf */ *maxbits = 0u; /* +0.0f */ }
}

// One wave per row, ROWS_PER_WAVE rows per wave, double-buffered async staging.
__global__ __launch_bounds__(256) void phase1_rows(
    const float* __restrict__ outputs, const int* __restrict__ targets,
    unsigned* __restrict__ counts, float* __restrict__ glp_partials,
    int B, int C)
{
    __shared__ __align__(16) float smem[WAVES_PER_BLOCK * 2 * LDS_ROW_FLOATS]; // 64 KB
    __shared__ float wave_glp[WAVES_PER_BLOCK];

    const int tid  = threadIdx.x;
    const int wave = tid >> 5;
    const int lane = tid & 31;

    float* buf0 = &smem[(wave * 2 + 0) * LDS_ROW_FLOATS];
    float* buf1 = &smem[(wave * 2 + 1) * LDS_ROW_FLOATS];

    const int rowBytes = C * 4;                    // 4000 for C=1000 (16B aligned)
    const int nChunks  = (rowBytes + 511) / 512;   // 512B per wave per async op

    const long long baseRow =
        ((long long)blockIdx.x * WAVES_PER_BLOCK + wave) * ROWS_PER_WAVE;

    auto issue_row = [&](int r, float* buf) {
        long long row = baseRow + r;
        if (row >= B) return;
        unsigned long long g = (unsigned long long)(const void*)(outputs + row * (long long)C)
                               + (unsigned)(lane * 16);
        unsigned l = lds_addr_u32(buf) + (unsigned)(lane * 16);
        for (int k = 0; k < nChunks; ++k) {
            if (k * 512 + lane * 16 < rowBytes)
                async_copy_b128(l + (unsigned)(k * 512), g + (unsigned)(k * 512));
        }
    };

    issue_row(0, buf0);
    float glp_acc = 0.0f;

    for (int r = 0; r < ROWS_PER_WAVE; ++r) {
        float* cur = (r & 1) ? buf1 : buf0;
        float* nxt = (r & 1) ? buf0 : buf1;

        if (r + 1 < ROWS_PER_WAVE) {
            issue_row(r + 1, nxt);
            if (nChunks == 8) wait_async_8();   // row r's 8 ops retired (in-order)
            else              wait_async_0();
        } else {
            wait_async_0();
        }

        long long row = baseRow + r;
        if (row < B) {
            // pass 1: per-lane max/argmax (ascending j => first-index tie-break)
            float vals[KMAX];
            float vmax = -__builtin_inff();
            int   vidx = C;
            #pragma unroll
            for (int k = 0; k < KMAX; ++k) {
                int j = lane + k * 32;
                float v = (j < C) ? cur[j] : -__builtin_inff();
                vals[k] = v;
                if (v > vmax) { vmax = v; vidx = j; }
            }
            // cross-lane argmax, ties -> lower index (matches jnp.argmax)
            for (int m = 16; m >= 1; m >>= 1) {
                float ov = __shfl_xor(vmax, m, 32);
                int   oi = __shfl_xor(vidx, m, 32);
                if (ov > vmax || (ov == vmax && oi < vidx)) { vmax = ov; vidx = oi; }
            }
            // pass 2: sum of exp via native v_exp_f32 (fixed order => deterministic)
            float se = 0.0f;
            #pragma unroll
            for (int k = 0; k < KMAX; ++k) {
                int j = lane + k * 32;
                if (j < C) se += __expf(vals[k] - vmax);
            }
            for (int m = 16; m >= 1; m >>= 1) se += __shfl_xor(se, m, 32);

            if (lane == 0) {
                int t = targets[row];
                glp_acc += cur[t] - vmax - __logf(se);
                atomicAdd(&counts[t * C + vidx], 1u);   // integer -> deterministic
            }
        }
    }

    if (lane == 0) wave_glp[wave] = glp_acc;
    __syncthreads();
    if (tid == 0) {
        float s = 0.0f;
        for (int w = 0; w < WAVES_PER_BLOCK; ++w) s += wave_glp[w];
        glp_partials[blockIdx.x] = s;
    }
}

// One block per cost-matrix row: val = (r==c)?1:cost+n+1 ; min/max + sum(n*val).
// No integer div/mod; rows are perfectly coalesced.
__global__ __launch_bounds__(256) void phase2_cm(
    const float* __restrict__ cost, const unsigned* __restrict__ counts,
    float* __restrict__ valsum_partials,
    unsigned* __restrict__ minbits, unsigned* __restrict__ maxbits, int C)
{
    const int tid = threadIdx.x;
    const int r   = blockIdx.x;
    const long long rowBase = (long long)r * C;

    float lmin = __builtin_inff(), lmax = 0.0f, lsum = 0.0f;
    for (int c = tid; c < C; c += 256) {
        long long idx = rowBase + c;
        unsigned n = counts[idx];
        float val = (c == r) ? 1.0f : (cost[idx] + (float)n + 1.0f);
        lmin = fminf(lmin, val);
        lmax = fmaxf(lmax, val);
        lsum += val * (float)n;
    }
    __shared__ float smin[256], smax[256], ssum[256];
    smin[tid] = lmin; smax[tid] = lmax; ssum[tid] = lsum;
    __syncthreads();
    for (int s = 128; s >= 1; s >>= 1) {
        if (tid < s) {
            smin[tid] = fminf(smin[tid], smin[tid + s]);
            smax[tid] = fmaxf(smax[tid], smax[tid + s]);
            ssum[tid] += ssum[tid + s];
        }
        __syncthreads();
    }
    if (tid == 0) {
        // all vals >= 1 > 0 => positive-float ordering == uint ordering
        atomicMin(minbits, __float_as_uint(smin[0]));
        atomicMax(maxbits, __float_as_uint(smax[0]));
        valsum_partials[r] = ssum[0];
    }
}

__global__ __launch_bounds__(256) void phase3_final(
    const float* __restrict__ glp_partials, int n1,
    const float* __restrict__ valsum_partials, int n2,
    const unsigned* __restrict__ minbits, const unsigned* __restrict__ maxbits,
    float* __restrict__ out, int B)
{
    __shared__ float s1[256], s2[256];
    const int tid = threadIdx.x;
    float a = 0.0f, b = 0.0f;
    for (int i = tid; i < n1; i += 256) a += glp_partials[i];
    for (int i = tid; i < n2; i += 256) b += valsum_partials[i];
    s1[tid] = a; s2[tid] = b;
    __syncthreads();
    for (int s = 128; s >= 1; s >>= 1) {
        if (tid < s) { s1[tid] += s1[tid + s]; s2[tid] += s2[tid + s]; }
        __syncthreads();
    }
    if (tid == 0) {
        const float mn = __uint_as_float(*minbits);
        const float mx = __uint_as_float(*maxbits);
        const float Bf = (float)B;
        const float sum_glp = s1[0];
        const float sum_val = s2[0];
        // gc_i = 1 + (val_i - mn)/(mx - mn);  sum_gc = B + (sum_val - B*mn)/(mx-mn)
        const float sum_gc = Bf + (sum_val - Bf * mn) / (mx - mn);
        out[0] = -((sum_glp / Bf) * (sum_gc / Bf));
    }
}

// ---------------------------------------------------------------------------
extern "C" void kernel_launch(void* const* d_in, const int* in_sizes, int n_in,
                              void* d_out, int out_size, void* d_ws, size_t ws_size,
                              hipStream_t stream) {
    const float* outputs = (const float*)d_in[0];   // [B, C] f32
    const int*   targets = (const int*)d_in[1];     // [B] i32
    const float* cost    = (const float*)d_in[2];   // [C, C] f32

    const int B = in_sizes[1];
    const int C = in_sizes[0] / in_sizes[1];
    const long long C2 = (long long)C * C;

    // workspace layout
    char* ws = (char*)d_ws;
    unsigned* counts = (unsigned*)ws;                                   // C2 u32
    long long off = (C2 * 4 + 255) & ~255LL;
    float* glp_partials = (float*)(ws + off);                           // blocks1 f32
    const int rowsPerBlock = WAVES_PER_BLOCK * ROWS_PER_WAVE;           // 64
    const int blocks1 = (B + rowsPerBlock - 1) / rowsPerBlock;          // 2048
    float* valsum_partials = glp_partials + blocks1;                    // C f32
    unsigned* minbits = (unsigned*)(valsum_partials + C);
    unsigned* maxbits = minbits + 1;

    const long long blocks0 = (C2 + 255) / 256;
    phase0_init<<<(unsigned)blocks0, 256, 0, stream>>>(counts, C2, minbits, maxbits);
    phase1_rows<<<blocks1, 256, 0, stream>>>(outputs, targets, counts, glp_partials, B, C);
    phase2_cm<<<C, 256, 0, stream>>>(cost, counts, valsum_partials, minbits, maxbits, C);
    phase3_final<<<1, 256, 0, stream>>>(glp_partials, blocks1, valsum_partials, C,
                                        minbits, maxbits, (float*)d_out, B);
}